// MinNormSolver_35880156791531
// MI455X (gfx1250) — compile-verified
//
#include <hip/hip_runtime.h>

typedef float v2f __attribute__((ext_vector_type(2)));
typedef float v8f __attribute__((ext_vector_type(8)));

#define N_TASKS   32
#define DIM       2097152
#define MAX_ITER  250
#define STOP_CRIT 1e-6f
#define EPSV      1e-8f

#define GRAM_BLOCKS       512
#define WAVES_PER_BLOCK   8
#define GRAM_THREADS      (WAVES_PER_BLOCK * 32)
#define TOTAL_WAVES       (GRAM_BLOCKS * WAVES_PER_BLOCK)
#define KSTEPS            (DIM / (4 * TOTAL_WAVES))   /* 128 ksteps of K=4 per wave */

static __device__ __forceinline__ v8f wmma_f32(v2f a, v2f b, v8f c) {
  // D = A(16x4,f32) * B(4x16,f32) + C(16x16,f32); 8-arg form:
  // (neg_a, A, neg_b, B, c_mod, C, reuse_a, reuse_b)
  return __builtin_amdgcn_wmma_f32_16x16x4_f32(false, a, false, b, (short)0, c,
                                               false, false);
}

// ---------------------------------------------------------------------------
// Kernel 1: G = V V^T via fp32 WMMA, streaming 268 MB from HBM exactly once.
// Each wave accumulates the full 32x32 G over its private D-slice, then the
// block reduces its 8 wave-partials in LDS and writes one deterministic
// partial G per block to workspace.
// ---------------------------------------------------------------------------
__global__ void __launch_bounds__(GRAM_THREADS)
gram_wmma(const float* __restrict__ V, float* __restrict__ partial) {
  __shared__ float lds[WAVES_PER_BLOCK * 1024];

  const int lane = threadIdx.x & 31;
  const int wv   = threadIdx.x >> 5;
  const long wave = (long)blockIdx.x * WAVES_PER_BLOCK + wv;

  const int h = lane >> 4;   // lane-half selects K pair {0,1} vs {2,3}
  const int r = lane & 15;   // lane selects M (A) / N (B) row of the block

  // Contiguous column slice for this wave; 8-byte aligned (colbase even).
  const long colbase = wave * (long)(KSTEPS * 4) + 2 * h;
  const float* p0 = V + (long)r        * DIM + colbase;  // rows 0..15
  const float* p1 = V + (long)(r + 16) * DIM + colbase;  // rows 16..31

  v8f acc00 = {}; v8f acc01 = {}; v8f acc10 = {}; v8f acc11 = {};

#pragma unroll 4
  for (int s = 0; s < KSTEPS; ++s) {
    v2f a0 = *(const v2f*)(p0 + 4 * s);   // A/B tile of row-block 0
    v2f a1 = *(const v2f*)(p1 + 4 * s);   // A/B tile of row-block 1
    acc00 = wmma_f32(a0, a0, acc00);      // G[ 0:16, 0:16]
    acc01 = wmma_f32(a0, a1, acc01);      // G[ 0:16,16:32]
    acc10 = wmma_f32(a1, a0, acc10);      // G[16:32, 0:16]
    acc11 = wmma_f32(a1, a1, acc11);      // G[16:32,16:32]
  }

  // C/D layout: acc[v] holds element (M = v + 8*h, N = lane&15) of its tile.
  float* base = lds + wv * 1024;
  const int c = lane & 15;
#pragma unroll
  for (int v = 0; v < 8; ++v) {
    const int m = v + 8 * h;
    base[(m)      * 32 + c]      = acc00[v];
    base[(m)      * 32 + 16 + c] = acc01[v];
    base[(m + 16) * 32 + c]      = acc10[v];
    base[(m + 16) * 32 + 16 + c] = acc11[v];
  }
  __syncthreads();

  for (int idx = threadIdx.x; idx < 1024; idx += GRAM_THREADS) {
    float s = 0.0f;
#pragma unroll
    for (int w = 0; w < WAVES_PER_BLOCK; ++w) s += lds[w * 1024 + idx];
    partial[(long)blockIdx.x * 1024 + idx] = s;
  }
}

// ---------------------------------------------------------------------------
// Wave32 helpers for the solver (one lane per task, n == warpSize == 32).
// ---------------------------------------------------------------------------
static __device__ __forceinline__ float wsum(float x) {
#pragma unroll
  for (int o = 16; o > 0; o >>= 1) x += __shfl_xor(x, o, 32);
  return x;
}
static __device__ __forceinline__ float wmin(float x) {
#pragma unroll
  for (int o = 16; o > 0; o >>= 1) x = fminf(x, __shfl_xor(x, o, 32));
  return x;
}
static __device__ __forceinline__ float matvec(const float* G, float x, int lane) {
  float acc = 0.0f;
#pragma unroll
  for (int j = 0; j < 32; ++j) acc = fmaf(G[lane * 32 + j], __shfl(x, j, 32), acc);
  return acc;
}
static __device__ __forceinline__ void line_solver(float v11, float v12, float v22,
                                                   float* gamma, float* cost) {
  float g = (v22 - v12) / (v11 + v22 - 2.0f * v12 + EPSV);
  float c = v22 + g * (v12 - v22);
  float gm = (v12 >= v22) ? 0.0f : g;
  float co = (v12 >= v22) ? v22  : c;
  if (v12 >= v11) { gm = 1.0f; co = v11; }
  *gamma = gm; *cost = co;
}

// Euclidean projection onto the simplex, per-lane value in, per-lane value out.
static __device__ __forceinline__ float proj_simplex(float v, int lane) {
  const float FINF = __builtin_inff();
  // Bitonic sort ascending across the wave.
  float x = v;
#pragma unroll
  for (int k = 2; k <= 32; k <<= 1) {
#pragma unroll
    for (int j = k >> 1; j > 0; j >>= 1) {
      float o = __shfl_xor(x, j, 32);
      bool up = ((lane & k) == 0);
      bool lo = ((lane & j) == 0);
      float mn = fminf(x, o), mx = fmaxf(x, o);
      x = (up == lo) ? mn : mx;
    }
  }
  float s = __shfl(x, 31 - lane, 32);     // descending order
  // Inclusive prefix sum of s.
  float cs = s;
#pragma unroll
  for (int o = 1; o < 32; o <<= 1) {
    float t = __shfl_up(cs, o, 32);
    if (lane >= o) cs += t;
  }
  float tmp_max = (cs - 1.0f) / (float)(lane + 1);
  float s_next = __shfl_down(s, 1, 32);
  bool cond = (lane < 31) && (tmp_max > s_next);
  unsigned long long b = __ballot(cond);
  unsigned int m = (unsigned int)b;
  int sel = (m != 0u) ? (__ffs(m) - 1) : 31;
  float tmax = __shfl(tmp_max, sel, 32);
  return fmaxf(v - tmax, 0.0f);
}

static __device__ __forceinline__ float next_point(const float* G, float cur,
                                                   float grad, int lane) {
  const float FINF = __builtin_inff();
  float proj = grad - wsum(grad) * (1.0f / 32.0f);
  bool neg = proj < 0.0f, pos = proj > 0.0f;
  float tm1 = neg ? (-cur / proj)         : FINF;
  float tm2 = pos ? ((1.0f - cur) / proj) : FINF;
  float m1 = wmin((tm1 > 1e-7f) ? tm1 : FINF);
  float t  = (m1 < 3.0e38f) ? m1 : 1.0f;
  float m2 = wmin((tm2 > 1e-7f) ? tm2 : FINF);
  t = fminf(t, m2);
  float nxt = fmaf(proj, t, cur);
  return proj_simplex(nxt, lane);
}

// ---------------------------------------------------------------------------
// Kernel 2: deterministically reduce block partials into G (LDS), then one
// wave32 runs planar init + 250 Frank-Wolfe iterations.
// ---------------------------------------------------------------------------
__global__ void __launch_bounds__(256)
solve_minnorm(const float* __restrict__ partial, float* __restrict__ out) {
  __shared__ float G[1024];
  const float FINF = __builtin_inff();

  for (int idx = threadIdx.x; idx < 1024; idx += 256) {
    float s = 0.0f;
    for (int b = 0; b < GRAM_BLOCKS; ++b) s += partial[(long)b * 1024 + idx];
    G[idx] = s;
  }
  __syncthreads();
  if (threadIdx.x >= 32) return;
  const int lane = threadIdx.x;

  // ---- planar init over the 496 upper-triangular pairs ----
  float bc = FINF, bg = 0.0f; int bp = 0x7fffffff;
  for (int p = lane; p < 496; p += 32) {
    int pp = p, i = 0, cnt = 31;
    while (pp >= cnt) { pp -= cnt; ++i; --cnt; }
    int j = i + 1 + pp;
    float gamma, cost;
    line_solver(G[i * 32 + i], G[i * 32 + j], G[j * 32 + j], &gamma, &cost);
    if (cost < bc) { bc = cost; bp = p; bg = gamma; }
  }
#pragma unroll
  for (int o = 16; o > 0; o >>= 1) {
    float oc = __shfl_xor(bc, o, 32);
    int   op = __shfl_xor(bp, o, 32);
    float og = __shfl_xor(bg, o, 32);
    if (oc < bc || (oc == bc && op < bp)) { bc = oc; bp = op; bg = og; }
  }
  int pp = bp, bi = 0, cnt = 31;
  while (pp >= cnt) { pp -= cnt; ++bi; --cnt; }
  int bj = bi + 1 + pp;
  float sol = (lane == bi) ? bg : ((lane == bj) ? (1.0f - bg) : 0.0f);

  // ---- Frank-Wolfe iterations ----
  for (int it = 0; it < MAX_ITER; ++it) {
    float Gs = matvec(G, sol, lane);
    float np = next_point(G, sol, -Gs, lane);
    float Gn = matvec(G, np, lane);
    float v11 = wsum(sol * Gs);
    float v12 = wsum(sol * Gn);
    float v22 = wsum(np * Gn);
    float gamma, cost;
    line_solver(v11, v12, v22, &gamma, &cost);
    float new_sol = gamma * sol + (1.0f - gamma) * np;
    float change = wsum(fabsf(new_sol - sol));
    if (change < STOP_CRIT) break;   // reference freezes sol once converged
    sol = new_sol;
  }
  out[lane] = sol;
}

extern "C" void kernel_launch(void* const* d_in, const int* in_sizes, int n_in,
                              void* d_out, int out_size, void* d_ws, size_t ws_size,
                              hipStream_t stream) {
  const float* V  = (const float*)d_in[0];
  float* out      = (float*)d_out;
  float* partials = (float*)d_ws;   // GRAM_BLOCKS * 1024 floats = 2 MB

  gram_wmma<<<GRAM_BLOCKS, GRAM_THREADS, 0, stream>>>(V, partials);
  solve_minnorm<<<1, 256, 0, stream>>>(partials, out);
}